// Wasserstein2dLoss_11072425689359
// MI455X (gfx1250) — compile-verified
//
#include <hip/hip_runtime.h>
#include <hip/hip_bf16.h>
#include <math.h>

// Problem constants (from the reference): B=8, C=4, N=32768, D=2, P=50
#define BB 8
#define CC 4
#define NN 32768
#define DD 2
#define PP 50
#define NT 1024   // threads per block = 32 waves (wave32)

typedef __attribute__((ext_vector_type(2))) float v2f;
typedef __attribute__((ext_vector_type(8))) float v8f;

// One workgroup per (b,c,p) slice. Projects both point sets onto direction p
// using V_WMMA_F32_16X16X4_F32 (K padded 2->4), bitonic-sorts both projected
// arrays entirely in LDS (2 x 128KB, fits CDNA5's 320KB/WGP), then reduces
// mean((sort(x)-sort(y))^2) into w2[bcp].
__global__ __launch_bounds__(NT, 1)
void swd_slice_kernel(const float* __restrict__ xpts,
                      const float* __restrict__ ypts,
                      const float* __restrict__ proj,
                      float* __restrict__ w2) {
  __shared__ float sx[NN];   // 128 KB
  __shared__ float sy[NN];   // 128 KB

  const int bcp = blockIdx.x;        // [0, B*C*P)
  const int p   = bcp % PP;
  const int bc  = bcp / PP;

  const float t0 = proj[p * DD + 0];
  const float t1 = proj[p * DD + 1];

  const int tid  = threadIdx.x;
  const int lane = tid & 31;
  const int wave = tid >> 5;               // 0..31
  const int nPerWave = NN / (NT / 32);     // 1024 rows per wave

  const float* __restrict__ xb = xpts + (size_t)bc * NN * DD;
  const float* __restrict__ yb = ypts + (size_t)bc * NN * DD;

  // ---- Projection via WMMA: D = A(16x4) * B(4x16), all B columns = theta ----
  // B layout (32-bit, 4x16): VGPR0 = row K=0 (lanes 0-15) / K=2 (lanes 16-31),
  //                          VGPR1 = row K=1 / K=3.  K=2,3 are the zero pad.
  v2f bmat;
  bmat[0] = (lane < 16) ? t0 : 0.0f;
  bmat[1] = (lane < 16) ? t1 : 0.0f;

  const int waveBase = wave * nPerWave;
  for (int base = waveBase; base < waveBase + nPerWave; base += 16) {
    // A layout (32-bit, 16x4): lanes 0-15 hold M=lane with VGPR0=K0, VGPR1=K1;
    // lanes 16-31 hold K=2,3 -> zero padding.
    v2f ax; ax[0] = 0.0f; ax[1] = 0.0f;
    v2f ay; ay[0] = 0.0f; ay[1] = 0.0f;
    if (lane < 16) {
      const float* px = xb + (size_t)(base + lane) * DD;
      ax[0] = px[0]; ax[1] = px[1];
      const float* py = yb + (size_t)(base + lane) * DD;
      ay[0] = py[0]; ay[1] = py[1];
    }
    v8f cz = {};
    v8f dx = __builtin_amdgcn_wmma_f32_16x16x4_f32(
        false, ax, false, bmat, (short)0, cz, false, false);
    v8f dy = __builtin_amdgcn_wmma_f32_16x16x4_f32(
        false, ay, false, bmat, (short)0, cz, false, false);
    // D layout: VGPR v -> row v (lanes 0-15) / row 8+v (lanes 16-31); every
    // column is identical, so column N==0 lanes scatter the 16 rows to LDS.
    if ((lane & 15) == 0) {
      const int r0 = base + ((lane >> 4) << 3);
#pragma unroll
      for (int v = 0; v < 8; ++v) {
        sx[r0 + v] = dx[v];
        sy[r0 + v] = dy[v];
      }
    }
  }

  // ---- In-LDS bitonic sort of both arrays (shared barriers) ----
  for (unsigned k = 2; k <= (unsigned)NN; k <<= 1) {
    for (unsigned j = k >> 1; j > 0; j >>= 1) {
      __syncthreads();
      for (unsigned i = tid; i < (unsigned)NN; i += NT) {
        const unsigned ixj = i ^ j;
        if (ixj > i) {
          const bool up = ((i & k) == 0);
          float a = sx[i], b = sx[ixj];
          if ((a > b) == up) { sx[i] = b; sx[ixj] = a; }
          float c = sy[i], d = sy[ixj];
          if ((c > d) == up) { sy[i] = d; sy[ixj] = c; }
        }
      }
    }
  }
  __syncthreads();

  // ---- mean of squared order-statistic differences ----
  float acc = 0.0f;
  for (unsigned i = tid; i < (unsigned)NN; i += NT) {
    const float dlt = sx[i] - sy[i];
    acc += dlt * dlt;
  }
#pragma unroll
  for (int off = 16; off > 0; off >>= 1)
    acc += __shfl_xor(acc, off, 32);
  __syncthreads();                 // done reading sx before reuse
  if (lane == 0) sx[wave] = acc;
  __syncthreads();
  if (tid == 0) {
    float tot = 0.0f;
    for (int w = 0; w < NT / 32; ++w) tot += sx[w];
    w2[bcp] = tot / (float)NN;     // W2^2 for this (b,c,p)
  }
}

// Tiny finalize: swd(b,c) = sqrt(mean_p w2), loss = mean_{b,c} swd
__global__ void swd_finalize_kernel(const float* __restrict__ w2,
                                    float* __restrict__ out) {
  __shared__ float sh[BB * CC];
  const int t = threadIdx.x;
  if (t < BB * CC) {
    float s = 0.0f;
    for (int p = 0; p < PP; ++p) s += w2[t * PP + p];
    sh[t] = sqrtf(s / (float)PP);
  }
  __syncthreads();
  if (t == 0) {
    float tot = 0.0f;
    for (int i = 0; i < BB * CC; ++i) tot += sh[i];
    out[0] = tot / (float)(BB * CC);
  }
}

extern "C" void kernel_launch(void* const* d_in, const int* in_sizes, int n_in,
                              void* d_out, int out_size, void* d_ws, size_t ws_size,
                              hipStream_t stream) {
  (void)in_sizes; (void)n_in; (void)out_size; (void)ws_size;
  const float* xpts = (const float*)d_in[0];   // output [B,C,N,D] f32
  const float* ypts = (const float*)d_in[1];   // target [B,C,N,D] f32
  const float* proj = (const float*)d_in[2];   // projections [P,D] f32
  float* w2 = (float*)d_ws;                    // 1600 floats of scratch

  swd_slice_kernel<<<BB * CC * PP, NT, 0, stream>>>(xpts, ypts, proj, w2);
  swd_finalize_kernel<<<1, 64, 0, stream>>>(w2, (float*)d_out);
}